// SpectreLayer_18743237279752
// MI455X (gfx1250) — compile-verified
//
#include <hip/hip_runtime.h>
#include <math.h>

// ---------------- problem constants ----------------
constexpr int B_ = 8;
constexpr int N_ = 4096;
constexpr int E_ = 1024;
constexpr int H_ = 16;
constexpr int D_ = 64;          // E/H
constexpr int NFFT = 4096;      // NMAX
constexpr int NF_ = NFFT / 2 + 1;   // 2049
constexpr int HID_ = 64;        // HIDDEN
constexpr float LN_EPS = 1e-5f;

// ---------------- WMMA helpers (CDNA5 / gfx1250, wave32) ----------------
typedef __attribute__((ext_vector_type(2))) float v2f;
typedef __attribute__((ext_vector_type(8))) float v8f;

// D(16x16,f32) = A(16x4,f32) x B(4x16,f32) + C   (codegen-confirmed round 0)
static __device__ inline v8f wmma_f32_16x16x4(v2f a, v2f b, v8f c) {
#if __has_builtin(__builtin_amdgcn_wmma_f32_16x16x4_f32)
  return __builtin_amdgcn_wmma_f32_16x16x4_f32(
      false, a, false, b, (short)0, c, false, false);
#else
  for (int i = 0; i < 8; ++i) c[i] += a.x * b.x + a.y * b.y;
  return c;
#endif
}

// ---------------- async LDS helpers (gfx1250, ASYNCcnt-tracked) -------------
static __device__ inline void async_ld_b32(unsigned lds_addr,
                                           unsigned long long gaddr) {
  asm volatile("global_load_async_to_lds_b32 %0, %1, off"
               :: "v"(lds_addr), "v"(gaddr) : "memory");
}
static __device__ inline void async_st_b32(unsigned long long gaddr,
                                           unsigned lds_addr) {
  asm volatile("global_store_async_from_lds_b32 %0, %1, off"
               :: "v"(gaddr), "v"(lds_addr) : "memory");
}
static __device__ inline void async_wait0() {
#if __has_builtin(__builtin_amdgcn_s_wait_asynccnt)
  __builtin_amdgcn_s_wait_asynccnt(0);
#else
  asm volatile("s_wait_asynccnt 0x0" ::: "memory");
#endif
}
static __device__ inline unsigned lds_off(const void* p) {
  return (unsigned)(uintptr_t)p;   // generic AS low 32 bits == LDS offset
}

// ---------------- kernel 1: column sums of x -> xmean (B,E) -----------------
__global__ __launch_bounds__(256) void k_xmean(const float* __restrict__ x,
                                               float* __restrict__ xmean) {
  int e  = blockIdx.x * 256 + threadIdx.x;
  int b  = blockIdx.y;
  int n0 = blockIdx.z * 256;
  const float* p = x + ((size_t)b * N_ + n0) * E_ + e;
  float s = 0.f;
#pragma unroll 8
  for (int i = 0; i < 256; ++i) s += p[(size_t)i * E_];
  atomicAdd(&xmean[b * E_ + e], s);
}

// ---------------- kernel 2: bar_q = (xmean/N) @ Wq^T  (B,E) -----------------
__global__ __launch_bounds__(256) void k_barq(const float* __restrict__ xmean,
                                              const float* __restrict__ Wq,
                                              float* __restrict__ barq) {
  int j = blockIdx.x * 256 + threadIdx.x;
  int b = j / E_, e = j % E_;
  const float* xm = xmean + b * E_;
  const float* wr = Wq + (size_t)e * E_;
  float s = 0.f;
#pragma unroll 8
  for (int k = 0; k < E_; ++k) s += xm[k] * wr[k];
  barq[j] = s * (1.0f / N_);
}

// ---------------- kernel 3: gate MLP (one WG per (b,h)) ---------------------
__global__ __launch_bounds__(256) void k_gate(const float* __restrict__ barq,
                                              const float* __restrict__ ln_g,
                                              const float* __restrict__ ln_b,
                                              const float* __restrict__ w1,
                                              const float* __restrict__ b1,
                                              const float* __restrict__ w2,
                                              const float* __restrict__ b2,
                                              float* __restrict__ gate) {
  __shared__ float qrow[D_];
  __shared__ float hbuf[HID_];
  __shared__ float red[2];
  int bh = blockIdx.x;
  int t  = threadIdx.x;

  if (t < D_) qrow[t] = barq[bh * D_ + t];
  __syncthreads();
  if (t == 0) {
    float mu = 0.f;
    for (int i = 0; i < D_; ++i) mu += qrow[i];
    mu *= (1.0f / D_);
    float var = 0.f;
    for (int i = 0; i < D_; ++i) { float d = qrow[i] - mu; var += d * d; }
    var *= (1.0f / D_);
    red[0] = mu;
    red[1] = rsqrtf(var + LN_EPS);
  }
  __syncthreads();
  if (t < D_) qrow[t] = (qrow[t] - red[0]) * red[1] * ln_g[t] + ln_b[t];
  __syncthreads();
  if (t < HID_) {
    float s = b1[t];
    const float* wr = w1 + (size_t)t * D_;
    for (int k = 0; k < D_; ++k) s += wr[k] * qrow[k];
    hbuf[t] = s / (1.0f + __expf(-s));   // silu
  }
  __syncthreads();
  for (int j = t; j < 2 * NF_; j += 256) {
    float s = b2[j];
    const float* wr = w2 + (size_t)j * HID_;
#pragma unroll 8
    for (int k = 0; k < HID_; ++k) s += wr[k] * hbuf[k];
    gate[(size_t)bh * 2 * NF_ + j] = s;
  }
}

// ------- kernel 4: v^T = (x @ Wv^T)^T stored (B,E,N); 2x2 WMMA tile/wave ----
__global__ __launch_bounds__(256) void k_vproj(const float* __restrict__ x,
                                               const float* __restrict__ Wv,
                                               float* __restrict__ vt) {
  int wave = threadIdx.x >> 5;
  int lane = threadIdx.x & 31;
  int we = wave >> 1;                       // 0..3
  int wn = wave & 1;                        // 0..1
  int e0 = blockIdx.x * 128 + we * 32;      // wave covers e' [e0, e0+32)
  int n0 = blockIdx.y * 64 + wn * 32;       // wave covers n  [n0, n0+32)
  int b  = blockIdx.z;
  int m  = lane & 15;
  int kk = (lane >> 4) << 1;

  const float* xb = x + (size_t)b * N_ * E_;
  const float* pa0 = xb + (size_t)(n0 + m) * E_;
  const float* pa1 = xb + (size_t)(n0 + 16 + m) * E_;
  const float* pb0 = Wv + (size_t)(e0 + m) * E_;
  const float* pb1 = Wv + (size_t)(e0 + 16 + m) * E_;

  v8f a00 = {}, a01 = {}, a10 = {}, a11 = {};
  for (int k0 = 0; k0 < E_; k0 += 32) {
    __builtin_prefetch(pa0 + k0 + 32, 0, 1);
    __builtin_prefetch(pa1 + k0 + 32, 0, 1);
    __builtin_prefetch(pb0 + k0 + 32, 0, 1);
    __builtin_prefetch(pb1 + k0 + 32, 0, 1);
#pragma unroll
    for (int kc = 0; kc < 32; kc += 4) {
      int k = k0 + kc + kk;
      v2f fa0, fa1, fb0, fb1;
      fa0.x = pa0[k]; fa0.y = pa0[k + 1];
      fa1.x = pa1[k]; fa1.y = pa1[k + 1];
      fb0.x = pb0[k]; fb0.y = pb0[k + 1];
      fb1.x = pb1[k]; fb1.y = pb1[k + 1];
      a00 = wmma_f32_16x16x4(fa0, fb0, a00);
      a01 = wmma_f32_16x16x4(fa0, fb1, a01);
      a10 = wmma_f32_16x16x4(fa1, fb0, a10);
      a11 = wmma_f32_16x16x4(fa1, fb1, a11);
    }
  }
  // transposed stores: lane owns column e' = e0+j*16+m,
  // rows n = n0 + i*16 + 8*(lane>>4) + r  -> 8 consecutive floats (2x b128)
  v8f accs[4] = {a00, a01, a10, a11};
#pragma unroll
  for (int i = 0; i < 2; ++i)
#pragma unroll
    for (int j = 0; j < 2; ++j) {
      v8f acc = accs[i * 2 + j];
      float* po = vt + ((size_t)b * E_ + (e0 + j * 16 + m)) * N_ +
                  n0 + i * 16 + ((lane >> 4) << 3);
      *reinterpret_cast<float4*>(po) =
          make_float4(acc[0], acc[1], acc[2], acc[3]);
      *reinterpret_cast<float4*>(po + 4) =
          make_float4(acc[4], acc[5], acc[6], acc[7]);
    }
}

// ---------------- kernel 5: FFT -> gate -> iFFT (in place on vt) ------------
static __device__ inline float2 cmul(float2 a, float2 b) {
  return make_float2(a.x * b.x - a.y * b.y, a.x * b.y + a.y * b.x);
}

static __device__ void fft_stockham(float2* __restrict__ x,
                                    float2* __restrict__ y,
                                    float sign, int t) {
  // radix-2 Stockham autosort: 12 stages, result ends in x.
  int n = NFFT, s = 1;
  float2* src = x;
  float2* dst = y;
  while (n > 1) {
    int m = n >> 1;
    float theta = sign * 6.28318530717958647692f / (float)n;
    for (int i = t; i < (NFFT >> 1); i += 256) {
      int p = i / s;
      int q = i - p * s;
      float sn, cs;
      __sincosf(theta * (float)p, &sn, &cs);
      float2 a = src[q + s * p];
      float2 b = src[q + s * (p + m)];
      dst[q + s * (2 * p)]     = make_float2(a.x + b.x, a.y + b.y);
      float2 dif = make_float2(a.x - b.x, a.y - b.y);
      dst[q + s * (2 * p + 1)] = make_float2(dif.x * cs - dif.y * sn,
                                             dif.x * sn + dif.y * cs);
    }
    __syncthreads();
    float2* tmp = src; src = dst; dst = tmp;
    n = m; s <<= 1;
  }
}

__global__ __launch_bounds__(256) void k_fft_gate(float* __restrict__ vt,
                                                  const float2* __restrict__ gate) {
  extern __shared__ float2 sm[];
  float2* A  = sm;
  float2* Bb = sm + NFFT;
  int t    = threadIdx.x;
  int pair = blockIdx.x;                  // D/2 = 32
  int bh   = blockIdx.y;                  // B*H = 128
  float* c0 = vt + ((size_t)bh * D_ + 2 * pair) * NFFT;  // (b*E + h*D + d)*N
  float* c1 = c0 + NFFT;

  // async gather of two real columns into interleaved complex LDS buffer
  for (int n = t; n < NFFT; n += 256) {
    unsigned l = lds_off(&A[n]);
    async_ld_b32(l,     (unsigned long long)(uintptr_t)(c0 + n));
    async_ld_b32(l + 4, (unsigned long long)(uintptr_t)(c1 + n));
  }
  async_wait0();
  __syncthreads();

  fft_stockham(A, Bb, -1.0f, t);          // forward; spectrum in A

  const float2* g = gate + (size_t)bh * NF_;
  for (int k = t; k <= NFFT / 2; k += 256) {
    float2 Zk = A[k];
    float2 Zn = A[(NFFT - k) & (NFFT - 1)];
    float2 V0 = make_float2(0.5f * (Zk.x + Zn.x), 0.5f * (Zk.y - Zn.y));
    float2 dd = make_float2(0.5f * (Zk.x - Zn.x), 0.5f * (Zk.y + Zn.y));
    float2 V1 = make_float2(dd.y, -dd.x);          // *(-i)
    float2 gk = g[k];
    float2 S0 = cmul(V0, gk);
    float2 S1 = cmul(V1, gk);
    if (k == 0 || k == NFFT / 2) {
      Bb[k] = make_float2(S0.x, S1.x);   // irfft drops imag at DC/Nyquist
    } else {
      Bb[k]        = make_float2(S0.x - S1.y,  S0.y + S1.x);  // S0 + i*S1
      Bb[NFFT - k] = make_float2(S0.x + S1.y, -S0.y + S1.x);  // conj pack
    }
  }
  __syncthreads();

  fft_stockham(Bb, A, 1.0f, t);           // inverse; result in Bb

  // scale in LDS, then async scatter back to the two real columns
  const float inv = 1.0f / (float)NFFT;
  for (int n = t; n < NFFT; n += 256) {
    float2 w = Bb[n];
    Bb[n] = make_float2(w.x * inv, w.y * inv);
  }
  __syncthreads();
  for (int n = t; n < NFFT; n += 256) {
    unsigned l = lds_off(&Bb[n]);
    async_st_b32((unsigned long long)(uintptr_t)(c0 + n), l);
    async_st_b32((unsigned long long)(uintptr_t)(c1 + n), l + 4);
  }
  async_wait0();
}

// ------- kernel 6: out = v-intermediate @ Wo^T; 2x2 WMMA tile/wave ----------
__global__ __launch_bounds__(256) void k_ogemm(const float* __restrict__ vt,
                                               const float* __restrict__ Wo,
                                               float* __restrict__ out) {
  int wave = threadIdx.x >> 5;
  int lane = threadIdx.x & 31;
  int we = wave >> 1;
  int wn = wave & 1;
  int e0 = blockIdx.x * 128 + we * 32;
  int n0 = blockIdx.y * 64 + wn * 32;
  int b  = blockIdx.z;
  int m  = lane & 15;
  int kk = (lane >> 4) << 1;

  const float* va0 = vt + (size_t)b * E_ * N_ + n0 + m;        // column-major in n
  const float* va1 = va0 + 16;
  const float* pb0 = Wo + (size_t)(e0 + m) * E_;
  const float* pb1 = Wo + (size_t)(e0 + 16 + m) * E_;

  v8f a00 = {}, a01 = {}, a10 = {}, a11 = {};
  for (int k0 = 0; k0 < E_; k0 += 32) {
    __builtin_prefetch(pb0 + k0 + 32, 0, 1);
    __builtin_prefetch(pb1 + k0 + 32, 0, 1);
#pragma unroll
    for (int kc = 0; kc < 32; kc += 4) {
      int k = k0 + kc + kk;
      const float* pa0 = va0 + (size_t)k * N_;
      const float* pa1 = va1 + (size_t)k * N_;
      v2f fa0, fa1, fb0, fb1;
      fa0.x = pa0[0]; fa0.y = pa0[N_];
      fa1.x = pa1[0]; fa1.y = pa1[N_];
      fb0.x = pb0[k]; fb0.y = pb0[k + 1];
      fb1.x = pb1[k]; fb1.y = pb1[k + 1];
      a00 = wmma_f32_16x16x4(fa0, fb0, a00);
      a01 = wmma_f32_16x16x4(fa0, fb1, a01);
      a10 = wmma_f32_16x16x4(fa1, fb0, a10);
      a11 = wmma_f32_16x16x4(fa1, fb1, a11);
    }
  }
  v8f accs[4] = {a00, a01, a10, a11};
#pragma unroll
  for (int i = 0; i < 2; ++i)
#pragma unroll
    for (int j = 0; j < 2; ++j) {
      v8f acc = accs[i * 2 + j];
      float* po = out + ((size_t)b * N_ + n0 + i * 16 + ((lane >> 4) << 3)) * E_ +
                  (e0 + j * 16 + m);
#pragma unroll
      for (int r = 0; r < 8; ++r) po[(size_t)r * E_] = acc[r];
    }
}

// ---------------- launcher ----------------
extern "C" void kernel_launch(void* const* d_in, const int* in_sizes, int n_in,
                              void* d_out, int out_size, void* d_ws, size_t ws_size,
                              hipStream_t stream) {
  const float* x    = (const float*)d_in[0];
  // d_in[1] = positions (int64): arange(N) with NMAX == N -> identity gather.
  const float* Wq   = (const float*)d_in[2];
  const float* Wv   = (const float*)d_in[3];
  const float* Wo   = (const float*)d_in[4];
  const float* ln_g = (const float*)d_in[5];
  const float* ln_b = (const float*)d_in[6];
  const float* w1   = (const float*)d_in[7];
  const float* b1   = (const float*)d_in[8];
  const float* w2   = (const float*)d_in[9];
  const float* b2   = (const float*)d_in[10];
  float* out = (float*)d_out;

  char* ws = (char*)d_ws;
  const size_t VT_BYTES = (size_t)B_ * E_ * N_ * sizeof(float);  // 128 MB
  float* vt    = (float*)(ws);
  float* xmean = (float*)(ws + VT_BYTES);
  float* barq  = (float*)(ws + VT_BYTES + 32768);
  float* gate  = (float*)(ws + VT_BYTES + 65536);

  hipMemsetAsync(xmean, 0, (size_t)B_ * E_ * sizeof(float), stream);

  k_xmean<<<dim3(E_ / 256, B_, N_ / 256), 256, 0, stream>>>(x, xmean);
  k_barq <<<dim3((B_ * E_) / 256), 256, 0, stream>>>(xmean, Wq, barq);
  k_gate <<<dim3(B_ * H_), 256, 0, stream>>>(barq, ln_g, ln_b, w1, b1, w2, b2, gate);
  k_vproj<<<dim3(E_ / 128, N_ / 64, B_), 256, 0, stream>>>(x, Wv, vt);
  k_fft_gate<<<dim3(D_ / 2, B_ * H_), 256, 2 * NFFT * sizeof(float2), stream>>>(
      vt, (const float2*)gate);
  k_ogemm<<<dim3(E_ / 128, N_ / 64, B_), 256, 0, stream>>>(vt, Wo, out);
}